// GeometricWorldModel_14834817040766
// MI455X (gfx1250) — compile-verified
//
#include <hip/hip_runtime.h>
#include <math.h>
#include <stdint.h>

#define ALPHA 0.1f

typedef __attribute__((ext_vector_type(2))) float v2f;
typedef __attribute__((ext_vector_type(8))) float v8f;

// D = A(16x4,f32) * B(4x16,f32) + C(16x16,f32), wave32, full-precision matrix pipe.
__device__ __forceinline__ v8f wmma_f32(v2f a, v2f b, v8f c) {
  return __builtin_amdgcn_wmma_f32_16x16x4_f32(
      /*neg_a=*/false, a, /*neg_b=*/false, b,
      /*c_mod=*/(short)0, c, /*reuse_a=*/false, /*reuse_b=*/false);
}

// ---------------- Kernel 1: H = (byte_embed[x] + pos_embed) @ Wp + bp -------
__global__ __launch_bounds__(32) void k_embed_gemm(
    const int* __restrict__ x, const float* __restrict__ byte_embed,
    const float* __restrict__ pos_embed, const float* __restrict__ Wp,
    const float* __restrict__ bp, float* __restrict__ H) {
  const int lane = threadIdx.x;
  const int half = lane >> 4;
  const int r    = lane & 15;
  const int n0 = blockIdx.x * 16;
  const int m0 = blockIdx.y * 16;
  const int row = m0 + r;         // 0..511 (= b*64 + t)
  const int t   = row & 63;
  const int tok = x[row];
  const float* be = byte_embed + tok * 256;
  const float* pe = pos_embed  + t   * 256;
  v8f acc = {};
  for (int kk = 0; kk < 64; ++kk) {
    const int k0 = kk * 4 + half * 2;
    v2f a, b;
    a.x = be[k0]     + pe[k0];
    a.y = be[k0 + 1] + pe[k0 + 1];
    b.x = Wp[k0 * 256 + n0 + r];
    b.y = Wp[(k0 + 1) * 256 + n0 + r];
    acc = wmma_f32(a, b, acc);
  }
  const float bias = bp[n0 + r];
#pragma unroll
  for (int j = 0; j < 8; ++j)
    H[(m0 + j + half * 8) * 256 + n0 + r] = acc[j] + bias;
}

// ---------------- Kernel 2: per-row norm + unbiased variance ----------------
__global__ __launch_bounds__(32) void k_rowstats(
    const float* __restrict__ H, float* __restrict__ invnrm,
    float* __restrict__ omega) {
  const int row = blockIdx.x;
  const int lane = threadIdx.x;
  const float* h = H + row * 256;
  float s = 0.f, sq = 0.f;
  for (int k = lane; k < 256; k += 32) { float v = h[k]; s += v; sq += v * v; }
  for (int off = 16; off > 0; off >>= 1) {
    s  += __shfl_xor(s,  off, 32);
    sq += __shfl_xor(sq, off, 32);
  }
  if (lane == 0) {
    const float mean = s * (1.f / 256.f);
    const float var  = (sq - 256.f * mean * mean) * (1.f / 255.f); // ddof=1
    omega[row]  = 1.f / (var + 1e-6f);
    invnrm[row] = 1.f / fmaxf(sqrtf(sq), 1e-12f);
  }
}

// ---------------- Kernel 3: per-batch Gram S = Hb @ Hb^T --------------------
__global__ __launch_bounds__(32) void k_gram(const float* __restrict__ H,
                                             float* __restrict__ S) {
  const int lane = threadIdx.x, half = lane >> 4, r = lane & 15;
  const int n0 = blockIdx.x * 16, m0 = blockIdx.y * 16, b = blockIdx.z;
  const float* Hb = H + b * 64 * 256;
  v8f acc = {};
  for (int kk = 0; kk < 64; ++kk) {
    const int k0 = kk * 4 + half * 2;
    v2f a, bb;
    a.x  = Hb[(m0 + r) * 256 + k0];
    a.y  = Hb[(m0 + r) * 256 + k0 + 1];
    bb.x = Hb[(n0 + r) * 256 + k0];      // B[k][n] = Hb[n][k]
    bb.y = Hb[(n0 + r) * 256 + k0 + 1];
    acc = wmma_f32(a, bb, acc);
  }
  float* Sb = S + b * 4096;
#pragma unroll
  for (int j = 0; j < 8; ++j) Sb[(m0 + j + half * 8) * 64 + n0 + r] = acc[j];
}

// ---------------- Kernel 4: A, degree, L_norm, K ----------------------------
__global__ __launch_bounds__(64) void k_laplacian(
    const float* __restrict__ S, const float* __restrict__ invnrm,
    const float* __restrict__ omega, const float* __restrict__ log_beta,
    float* __restrict__ Lnorm_out, float* __restrict__ Kmat) {
  __shared__ float inv_s[64];
  __shared__ float dis_s[64];
  __shared__ float A_s[64 * 64];
  const int b = blockIdx.x, i = threadIdx.x;
  inv_s[i] = invnrm[b * 64 + i];
  __syncthreads();
  const float* Sb = S + b * 4096;
  const float inv_i = inv_s[i];
  float dv = 0.f;
  for (int j = 0; j < 64; ++j) {
    float a = (i == j) ? 0.f : fmaxf(Sb[i * 64 + j], 0.f) * inv_i * inv_s[j];
    A_s[i * 64 + j] = a;
    dv += a;
  }
  dis_s[i] = 1.f / (sqrtf(dv) + 1e-6f);
  __syncthreads();
  const float beta = expf(log_beta[0]);
  const float om = omega[b * 64 + i];
  const float di = dis_s[i];
  for (int j = 0; j < 64; ++j) {
    float l  = ((i == j) ? dv : 0.f) - A_s[i * 64 + j];
    float ln = di * l * dis_s[j];
    Lnorm_out[b * 4096 + i * 64 + j] = ln;
    Kmat[b * 4096 + i * 64 + j] = ((i == j) ? om : 0.f) + beta * ln;
  }
}

// ---------------- Kernel 5: parallel-ordered Jacobi eigensolver -------------
// Round-robin tournament: each round rotates 32 disjoint (p,q) pairs at once.
// 63 rounds/sweep, 3 barriers/round, 256 threads. K staged into LDS with the
// CDNA5 async global->LDS path (ASYNCcnt).
__global__ __launch_bounds__(256) void k_jacobi(const float* __restrict__ Kin,
                                                float* __restrict__ Lam,
                                                float* __restrict__ PhiOut) {
  __shared__ __align__(16) float Km[64 * 64];
  __shared__ __align__(16) float Ph[64 * 64];
  __shared__ float cs_c[32], cs_s[32];
  __shared__ int   pr_p[32], pr_q[32];
  __shared__ float lam_s[64];
  __shared__ int   idx_s[64];
  const int b = blockIdx.x, tid = threadIdx.x;

  // --- async-load the 16KB K matrix into LDS (b128: 4 floats/lane/op) ------
  {
    const float* src = Kin + b * 4096;   // uniform -> SGPR pair
#pragma unroll
    for (int it = 0; it < 4; ++it) {
      const int elem = (it * 256 + tid) * 4;                 // float index
      unsigned lds_addr = (unsigned)(uintptr_t)(&Km[elem]);  // LDS byte offset
      unsigned voff = (unsigned)(elem * 4);                  // global byte off
      asm volatile("global_load_async_to_lds_b128 %0, %1, %2"
                   :: "v"(lds_addr), "v"(voff), "s"(src) : "memory");
    }
    asm volatile("s_wait_asynccnt 0x0" ::: "memory");
  }
  for (int j = tid; j < 4096; j += 256)
    Ph[j] = ((j >> 6) == (j & 63)) ? 1.f : 0.f;
  __syncthreads();

  for (int sweep = 0; sweep < 12; ++sweep) {
    for (int r = 0; r < 63; ++r) {
      // phase 1: 32 rotations from the pre-round matrix
      if (tid < 32) {
        int p, q;
        if (tid == 0) { p = 63; q = r % 63; }
        else {
          p = (r + tid) % 63;
          q = (r - tid + 63) % 63;
        }
        if (p > q) { int tswp = p; p = q; q = tswp; }
        pr_p[tid] = p; pr_q[tid] = q;
        const float apq = Km[p * 64 + q];
        float c = 1.f, s = 0.f;
        if (fabsf(apq) > 1e-12f) {
          const float app = Km[p * 64 + p], aqq = Km[q * 64 + q];
          const float tau = (aqq - app) / (2.f * apq);
          const float tt  = ((tau >= 0.f) ? 1.f : -1.f) /
                            (fabsf(tau) + sqrtf(1.f + tau * tau));
          c = rsqrtf(1.f + tt * tt);
          s = tt * c;
        }
        cs_c[tid] = c; cs_s[tid] = s;
      }
      __syncthreads();
      // phase 2: column rotations of K and Phi (pairs disjoint -> no conflicts)
      {
        const int j = tid & 31;
        const int p = pr_p[j], q = pr_q[j];
        const float c = cs_c[j], s = cs_s[j];
        for (int i = tid >> 5; i < 64; i += 8) {
          const float kip = Km[i * 64 + p], kiq = Km[i * 64 + q];
          Km[i * 64 + p] = c * kip - s * kiq;
          Km[i * 64 + q] = s * kip + c * kiq;
          const float pip = Ph[i * 64 + p], piq = Ph[i * 64 + q];
          Ph[i * 64 + p] = c * pip - s * piq;
          Ph[i * 64 + q] = s * pip + c * piq;
        }
      }
      __syncthreads();
      // phase 3: row rotations of K
      {
        const int j = tid & 31;
        const int p = pr_p[j], q = pr_q[j];
        const float c = cs_c[j], s = cs_s[j];
        for (int i = tid >> 5; i < 64; i += 8) {
          const float kpi = Km[p * 64 + i], kqi = Km[q * 64 + i];
          Km[p * 64 + i] = c * kpi - s * kqi;
          Km[q * 64 + i] = s * kpi + c * kqi;
        }
      }
      __syncthreads();
    }
  }

  if (tid < 64) { lam_s[tid] = Km[tid * 64 + tid]; idx_s[tid] = tid; }
  __syncthreads();
  if (tid == 0) {  // ascending insertion sort with index permutation
    for (int a = 1; a < 64; ++a) {
      const float key = lam_s[a];
      const int ki = idx_s[a];
      int w = a - 1;
      while (w >= 0 && lam_s[w] > key) {
        lam_s[w + 1] = lam_s[w];
        idx_s[w + 1] = idx_s[w];
        --w;
      }
      lam_s[w + 1] = key;
      idx_s[w + 1] = ki;
    }
  }
  __syncthreads();
  if (tid < 64) Lam[b * 64 + tid] = fmaxf(lam_s[tid], 1e-6f);
  for (int idx = tid; idx < 4096; idx += 256) {
    const int i = idx >> 6, k = idx & 63;
    PhiOut[b * 4096 + idx] = Ph[i * 64 + idx_s[k]];
  }
}

// ---------------- Kernel 6: z = Phi^T @ H (per batch) -----------------------
__global__ __launch_bounds__(32) void k_z(const float* __restrict__ Phi,
                                          const float* __restrict__ H,
                                          float* __restrict__ z_out) {
  const int lane = threadIdx.x, half = lane >> 4, r = lane & 15;
  const int n0 = blockIdx.x * 16, m0 = blockIdx.y * 16, b = blockIdx.z;
  const float* Pb = Phi + b * 4096;
  const float* Hb = H + b * 64 * 256;
  v8f acc = {};
  for (int kk = 0; kk < 16; ++kk) {
    const int k0 = kk * 4 + half * 2;
    v2f a, bb;
    a.x  = Pb[k0 * 64 + m0 + r];        // A[m][k] = Phi[k][m]
    a.y  = Pb[(k0 + 1) * 64 + m0 + r];
    bb.x = Hb[k0 * 256 + n0 + r];
    bb.y = Hb[(k0 + 1) * 256 + n0 + r];
    acc = wmma_f32(a, bb, acc);
  }
#pragma unroll
  for (int j = 0; j < 8; ++j)
    z_out[(b * 64 + m0 + j + half * 8) * 256 + n0 + r] = acc[j];
}

// ---------------- Kernel 7: fused z-mean / ||z_mean||^2 / g (per batch) -----
__global__ __launch_bounds__(256) void k_stats_g(const float* __restrict__ z,
                                                 const float* __restrict__ Lam,
                                                 float* __restrict__ g) {
  __shared__ float zm[64];
  __shared__ float red;
  const int b = blockIdx.x, tid = threadIdx.x;
  const int wave = tid >> 5, lane = tid & 31;
  for (int t = wave; t < 64; t += 8) {
    const float* zr = z + (b * 64 + t) * 256;
    float s = 0.f;
    for (int k = lane; k < 256; k += 32) s += zr[k];
    for (int off = 16; off > 0; off >>= 1) s += __shfl_xor(s, off, 32);
    if (lane == 0) zm[t] = s * (1.f / 256.f);
  }
  __syncthreads();
  if (wave == 0) {
    float s = zm[lane] * zm[lane] + zm[lane + 32] * zm[lane + 32];
    for (int off = 16; off > 0; off >>= 1) s += __shfl_xor(s, off, 32);
    if (lane == 0) red = s;
  }
  __syncthreads();
  const float znsq = red;
  for (int idx = tid; idx < 4096; idx += 256) {
    const int i = idx >> 6, j = idx & 63;
    float val = 2.f * ALPHA * zm[i] * zm[j];
    if (i == j) val += Lam[b * 64 + i] + ALPHA * znsq;
    g[b * 4096 + idx] = val;
  }
}

// ---------------- Kernel 8: v = temporal diff of z (float4 stores) ----------
__global__ void k_v(const float4* __restrict__ z4, float4* __restrict__ v4) {
  const int idx = blockIdx.x * blockDim.x + threadIdx.x;  // 32768 float4
  if (idx >= 8 * 64 * 64) return;
  const int t = (idx >> 6) & 63;   // 64 float4 per (b,t) row
  float4 out = {0.f, 0.f, 0.f, 0.f};
  if (t != 0) {
    const float4 a = z4[idx], bq = z4[idx - 64];
    out.x = a.x - bq.x; out.y = a.y - bq.y;
    out.z = a.z - bq.z; out.w = a.w - bq.w;
  }
  v4[idx] = out;
}

// ---------------- Kernel 9: Gamma (diag scale, float4 b128 stores) ----------
__global__ void k_gamma(const float* __restrict__ Lam, float4* __restrict__ G4) {
  const int idx = blockIdx.x * blockDim.x + threadIdx.x;  // 524288 float4
  if (idx >= 8 * 64 * 64 * 16) return;
  const int j4 = (idx & 15) * 4;
  const int i  = (idx >> 4) & 63;
  const int k  = (idx >> 10) & 63;
  const int b  = idx >> 16;
  const float dv = ALPHA / (Lam[b * 64 + k] + 1e-6f);
  float4 out;
  out.x = (i == j4)     ? dv : 0.f;
  out.y = (i == j4 + 1) ? dv : 0.f;
  out.z = (i == j4 + 2) ? dv : 0.f;
  out.w = (i == j4 + 3) ? dv : 0.f;
  G4[idx] = out;
}

// ---------------- Kernel 10: logits = Z @ Wd + bd ---------------------------
__global__ __launch_bounds__(32) void k_logits(const float* __restrict__ Z,
                                               const float* __restrict__ Wd,
                                               const float* __restrict__ bd,
                                               float* __restrict__ out) {
  const int lane = threadIdx.x, half = lane >> 4, r = lane & 15;
  const int n0 = blockIdx.x * 16, m0 = blockIdx.y * 16;
  const float* arow = Z + (m0 + r) * 256;
  v8f acc = {};
  for (int kk = 0; kk < 64; ++kk) {
    const int k0 = kk * 4 + half * 2;
    v2f a, b;
    a.x = arow[k0];
    a.y = arow[k0 + 1];
    b.x = Wd[k0 * 256 + n0 + r];
    b.y = Wd[(k0 + 1) * 256 + n0 + r];
    acc = wmma_f32(a, b, acc);
  }
  const float bias = bd[n0 + r];
#pragma unroll
  for (int j = 0; j < 8; ++j)
    out[(m0 + j + half * 8) * 256 + n0 + r] = acc[j] + bias;
}

// ---------------------------------------------------------------------------
extern "C" void kernel_launch(void* const* d_in, const int* in_sizes, int n_in,
                              void* d_out, int out_size, void* d_ws,
                              size_t ws_size, hipStream_t stream) {
  (void)in_sizes; (void)n_in; (void)out_size; (void)ws_size;
  const int*   x          = (const int*)d_in[0];
  const float* byte_embed = (const float*)d_in[1];
  const float* pos_embed  = (const float*)d_in[2];
  const float* Wp         = (const float*)d_in[3];
  const float* bp         = (const float*)d_in[4];
  const float* log_beta   = (const float*)d_in[5];
  const float* Wd         = (const float*)d_in[6];
  const float* bd         = (const float*)d_in[7];

  float* out = (float*)d_out;
  float* out_logits = out;                 // 131072
  float* out_z      = out + 131072;        // 131072
  float* out_v      = out + 262144;        // 131072
  float* out_g      = out + 393216;        // 32768
  float* out_Gamma  = out + 425984;        // 2097152
  float* out_Lnorm  = out + 2523136;       // 32768

  float* ws = (float*)d_ws;
  float* H      = ws;                      // 131072
  float* S      = ws + 131072;             // 32768
  float* invnrm = ws + 163840;             // 512
  float* omega  = ws + 164352;             // 512
  float* Kmat   = ws + 164864;             // 32768
  float* Lam    = ws + 197632;             // 512
  float* Phi    = ws + 198144;             // 32768

  k_embed_gemm<<<dim3(16, 32), 32, 0, stream>>>(x, byte_embed, pos_embed, Wp,
                                                bp, H);
  k_rowstats<<<512, 32, 0, stream>>>(H, invnrm, omega);
  k_gram<<<dim3(4, 4, 8), 32, 0, stream>>>(H, S);
  k_laplacian<<<8, 64, 0, stream>>>(S, invnrm, omega, log_beta, out_Lnorm,
                                    Kmat);
  k_jacobi<<<8, 256, 0, stream>>>(Kmat, Lam, Phi);
  k_z<<<dim3(16, 4, 8), 32, 0, stream>>>(Phi, H, out_z);
  k_stats_g<<<8, 256, 0, stream>>>(out_z, Lam, out_g);
  k_v<<<(8 * 64 * 64 + 255) / 256, 256, 0, stream>>>((const float4*)out_z,
                                                     (float4*)out_v);
  k_gamma<<<(8 * 64 * 64 * 16 + 255) / 256, 256, 0, stream>>>(
      Lam, (float4*)out_Gamma);
  k_logits<<<dim3(16, 32), 32, 0, stream>>>(out_z, Wd, bd, out_logits);
}